// GAT_22711787061921
// MI455X (gfx1250) — compile-verified
//
#include <hip/hip_runtime.h>
#include <math.h>

typedef __attribute__((ext_vector_type(2))) float v2f;
typedef __attribute__((ext_vector_type(8))) float v8f;

#define NEG_SLOPE 0.2f
#define ORD_NEG_INF 0x007FFFFFu  // ordered encoding of -inf

__device__ __forceinline__ unsigned int ford(float f) {
    unsigned int u = __float_as_uint(f);
    return (u & 0x80000000u) ? ~u : (u | 0x80000000u);
}
__device__ __forceinline__ float funord(unsigned int u) {
    u = (u & 0x80000000u) ? (u & 0x7FFFFFFFu) : ~u;
    return __uint_as_float(u);
}

// ---------------------------------------------------------------------------
// H[Nn,F] = X[Nn,K] @ W[K,F]  via V_WMMA_F32_16X16X4_F32, K/F compile-time.
// One wave per 16x16 output tile; fully unrolled chain of 16x16x4 WMMAs.
// A layout (16x4 fp32): lanes 0-15 -> M=lane, K=(k0,k0+1); lanes 16-31 -> K=(k0+2,k0+3)
// B layout (4x16 fp32): mirrored (row K striped across lanes within a VGPR)
// C/D: 8 VGPRs, N = lane&15, M = r + 8*(lane>>4)
// Nn is a multiple of 16 (50000 = 3125*16); row clamp keeps loads in range
// without EXEC predication so the WMMA path stays branch-free.
// ---------------------------------------------------------------------------
template <int K, int F>
__global__ void gat_gemm_wmma_f32(const float* __restrict__ X,
                                  const float* __restrict__ W,
                                  float* __restrict__ H, int Nn) {
    const int wave = threadIdx.x >> 5;
    const int lane = threadIdx.x & 31;
    const int mt = blockIdx.x * (blockDim.x >> 5) + wave;   // uniform per wave
    const int nt = blockIdx.y;
    if (mt * 16 >= Nn) return;                               // whole-wave exit

    int row = mt * 16 + (lane & 15);
    row = row < Nn ? row : Nn - 1;                           // clamp, no divergence
    const int col   = nt * 16 + (lane & 15);
    const int khalf = (lane >> 4) * 2;                       // 0 or 2

    const float* __restrict__ pA = X + (long long)row * K + khalf;
    const float* __restrict__ pB = W + (long long)khalf * F + col;

    v8f acc = {};
#pragma unroll
    for (int k0 = 0; k0 < K; k0 += 4) {
        v2f a, b;
        a = *(const v2f*)(pA + k0);                          // contiguous -> b64 load
        b.x = pB[(long long)k0 * F];
        b.y = pB[(long long)(k0 + 1) * F];
        acc = __builtin_amdgcn_wmma_f32_16x16x4_f32(
            /*neg_a=*/false, a, /*neg_b=*/false, b,
            /*c_mod=*/(short)0, acc, /*reuse_a=*/false, /*reuse_b=*/false);
    }

    const int rbase = mt * 16 + (lane >> 4) * 8;
    float* __restrict__ pH = H + (long long)rbase * F + col;
#pragma unroll
    for (int r = 0; r < 8; ++r)
        if (rbase + r < Nn) pH[(long long)r * F] = acc[r];
}

// as_n = H @ a_src, ad_n = H @ a_dst   (one wave per node, lane-parallel over F)
__global__ void gat_node_logits(const float* __restrict__ H,
                                const float* __restrict__ a_src,
                                const float* __restrict__ a_dst,
                                float* __restrict__ as_n, float* __restrict__ ad_n,
                                int Nn, int F) {
    const int wave = threadIdx.x >> 5;
    const int lane = threadIdx.x & 31;
    const int n = blockIdx.x * (blockDim.x >> 5) + wave;
    if (n >= Nn) return;
    float s0 = 0.f, s1 = 0.f;
    for (int f = lane; f < F; f += 32) {
        float h = H[(long long)n * F + f];
        s0 += h * a_src[f];
        s1 += h * a_dst[f];
    }
#pragma unroll
    for (int off = 16; off > 0; off >>= 1) {
        s0 += __shfl_xor(s0, off, 32);
        s1 += __shfl_xor(s1, off, 32);
    }
    if (lane == 0) { as_n[n] = s0; ad_n[n] = s1; }
}

__global__ void gat_fill_u32(unsigned int* __restrict__ p, unsigned int v, int n) {
    int i = blockIdx.x * blockDim.x + threadIdx.x;
    if (i < n) p[i] = v;
}

// pass 1: e = leaky_relu(as[src] + ad[dst]); segment max over dst (ordered-uint atomicMax)
__global__ void gat_edge_max(const int* __restrict__ ei, int E, int Nn,
                             const float* __restrict__ as_n, const float* __restrict__ ad_n,
                             float* __restrict__ eval, unsigned int* __restrict__ m_ord) {
    int e = blockIdx.x * blockDim.x + threadIdx.x;
    const int tot = E + Nn;                       // self loops appended
    if (e >= tot) return;
    int src, dst;
    if (e < E) { src = ei[e]; dst = ei[E + e]; } else { src = dst = e - E; }
    float v = as_n[src] + ad_n[dst];
    v = (v >= 0.f) ? v : NEG_SLOPE * v;
    eval[e] = v;
    atomicMax(&m_ord[dst], ford(v));
}

// pass 2: ee = exp(e - m[dst]); segment sum over dst
__global__ void gat_edge_expsum(const int* __restrict__ ei, int E, int Nn,
                                const unsigned int* __restrict__ m_ord,
                                float* __restrict__ eval, float* __restrict__ s) {
    int e = blockIdx.x * blockDim.x + threadIdx.x;
    const int tot = E + Nn;
    if (e >= tot) return;
    int dst = (e < E) ? ei[E + e] : (e - E);
    float m  = funord(m_ord[dst]);
    float ee = __expf(eval[e] - m);
    eval[e] = ee;
    atomicAdd(&s[dst], ee);
}

// pass 3: out[dst] += alpha * h[src]   (wave per edge, lanes over features)
__global__ void gat_edge_scatter(const int* __restrict__ ei, int E, int Nn, int F,
                                 const float* __restrict__ eval, const float* __restrict__ s,
                                 const float* __restrict__ H, float* __restrict__ out) {
    const int wave = threadIdx.x >> 5;
    const int lane = threadIdx.x & 31;
    const int e = blockIdx.x * (blockDim.x >> 5) + wave;
    const int tot = E + Nn;
    if (e >= tot) return;
    int src, dst;
    if (e < E) { src = ei[e]; dst = ei[E + e]; } else { src = dst = e - E; }
    float alpha = eval[e] / (s[dst] + 1e-16f);
    for (int f = lane; f < F; f += 32)
        atomicAdd(&out[(long long)dst * F + f], alpha * H[(long long)src * F + f]);
}

__global__ void gat_bias_act(float* __restrict__ out, const float* __restrict__ b,
                             long long total, int F, int do_relu) {
    long long i = (long long)blockIdx.x * blockDim.x + threadIdx.x;
    if (i >= total) return;
    float v = out[i] + b[(int)(i % F)];
    if (do_relu) v = v > 0.f ? v : 0.f;
    out[i] = v;
}

// global_add_pool: g[batch[n], f] += h[n, f]
__global__ void gat_pool(const float* __restrict__ h, const int* __restrict__ batch,
                         float* __restrict__ g, int Nn, int F) {
    long long i = (long long)blockIdx.x * blockDim.x + threadIdx.x;
    if (i >= (long long)Nn * F) return;
    int n = (int)(i / F), f = (int)(i % F);
    atomicAdd(&g[batch[n] * F + f], h[i]);
}

__global__ void gat_mlp(const float* __restrict__ in, const float* __restrict__ W,
                        const float* __restrict__ b, float* __restrict__ out,
                        int M, int K, int Ncol, int do_relu) {
    int i = blockIdx.x * blockDim.x + threadIdx.x;
    if (i >= M * Ncol) return;
    int m = i / Ncol, c = i % Ncol;
    float acc = b[c];
    for (int k = 0; k < K; ++k) acc += in[m * K + k] * W[k * Ncol + c];
    if (do_relu && acc < 0.f) acc = 0.f;
    out[i] = acc;
}

// log_softmax over axis 0 (per class column across G graphs)
__global__ void gat_log_softmax0(const float* __restrict__ logits, float* __restrict__ out,
                                 int G, int C) {
    __shared__ float lse[32];
    int tid = threadIdx.x;
    if (tid < C) {
        float mx = -INFINITY;
        for (int g = 0; g < G; ++g) mx = fmaxf(mx, logits[g * C + tid]);
        float sum = 0.f;
        for (int g = 0; g < G; ++g) sum += __expf(logits[g * C + tid] - mx);
        lse[tid] = mx + __logf(sum);
    }
    __syncthreads();
    for (int i = tid; i < G * C; i += blockDim.x)
        out[i] = logits[i] - lse[i % C];
}

// ---------------------------------------------------------------------------
extern "C" void kernel_launch(void* const* d_in, const int* in_sizes, int n_in,
                              void* d_out, int out_size, void* d_ws, size_t ws_size,
                              hipStream_t stream) {
    const float* x     = (const float*)d_in[0];
    const int*   ei    = (const int*)d_in[1];   // [2, E] int32 (jnp int64 aliases int32, x64 off)
    const int*   batch = (const int*)d_in[2];
    const float* W[3]     = {(const float*)d_in[3], (const float*)d_in[7],  (const float*)d_in[11]};
    const float* a_src[3] = {(const float*)d_in[4], (const float*)d_in[8],  (const float*)d_in[12]};
    const float* a_dst[3] = {(const float*)d_in[5], (const float*)d_in[9],  (const float*)d_in[13]};
    const float* bias[3]  = {(const float*)d_in[6], (const float*)d_in[10], (const float*)d_in[14]};
    const float* lin1_W = (const float*)d_in[15];
    const float* lin1_b = (const float*)d_in[16];
    const float* lin2_W = (const float*)d_in[17];
    const float* lin2_b = (const float*)d_in[18];

    const int Fout[3] = {32, 64, 128};
    const int Nn = in_sizes[0] / 128;
    const int E  = in_sizes[1] / 2;
    const int G  = 64, C = 10;
    const int tot = E + Nn;

    // workspace layout (floats)
    float* ws = (float*)d_ws;
    size_t off = 0;
    float* H    = ws + off; off += (size_t)Nn * 128;     // GEMM result (max F)
    float* O[3];
    O[0] = ws + off; off += (size_t)Nn * 32;
    O[1] = ws + off; off += (size_t)Nn * 64;
    O[2] = ws + off; off += (size_t)Nn * 128;
    float* as_n = ws + off; off += Nn;
    float* ad_n = ws + off; off += Nn;
    unsigned int* m_ord = (unsigned int*)(ws + off); off += Nn;
    float* ssum = ws + off; off += Nn;
    float* eval = ws + off; off += tot;
    float* g    = ws + off; off += (size_t)G * 128;
    float* g1   = ws + off; off += (size_t)G * 32;
    float* lg   = ws + off; off += (size_t)G * C;
    (void)ws_size; (void)n_in; (void)out_size;

    const int mtiles = (Nn + 15) / 16;
    const int gx = (mtiles + 3) / 4;                     // 4 waves per block

    const float* in_feat = x;
    for (int L = 0; L < 3; ++L) {
        const int F = Fout[L];
        // 1) H = in_feat @ W[L]   (fp32 WMMA, one 16x16 tile per wave)
        switch (L) {
        case 0: gat_gemm_wmma_f32<128, 32><<<dim3(gx, 2), 128, 0, stream>>>(in_feat, W[0], H, Nn); break;
        case 1: gat_gemm_wmma_f32< 32, 64><<<dim3(gx, 4), 128, 0, stream>>>(in_feat, W[1], H, Nn); break;
        case 2: gat_gemm_wmma_f32< 64,128><<<dim3(gx, 8), 128, 0, stream>>>(in_feat, W[2], H, Nn); break;
        }
        // 2) per-node attention logits
        gat_node_logits<<<(Nn + 3) / 4, 128, 0, stream>>>(H, a_src[L], a_dst[L], as_n, ad_n, Nn, F);
        // 3) init segment state + output accumulator
        gat_fill_u32<<<(Nn + 255) / 256, 256, 0, stream>>>(m_ord, ORD_NEG_INF, Nn);
        hipMemsetAsync(ssum, 0, (size_t)Nn * sizeof(float), stream);
        hipMemsetAsync(O[L], 0, (size_t)Nn * F * sizeof(float), stream);
        // 4) segment softmax + weighted scatter
        gat_edge_max<<<(tot + 255) / 256, 256, 0, stream>>>(ei, E, Nn, as_n, ad_n, eval, m_ord);
        gat_edge_expsum<<<(tot + 255) / 256, 256, 0, stream>>>(ei, E, Nn, m_ord, eval, ssum);
        gat_edge_scatter<<<(tot + 7) / 8, 256, 0, stream>>>(ei, E, Nn, F, eval, ssum, H, O[L]);
        // 5) bias (+ relu on layers 1,2)
        long long totNF = (long long)Nn * F;
        gat_bias_act<<<(int)((totNF + 255) / 256), 256, 0, stream>>>(O[L], bias[L], totNF, F, L < 2 ? 1 : 0);
        in_feat = O[L];
    }

    // global_add_pool + MLP head + column log_softmax
    hipMemsetAsync(g, 0, (size_t)G * 128 * sizeof(float), stream);
    long long np = (long long)Nn * 128;
    gat_pool<<<(int)((np + 255) / 256), 256, 0, stream>>>(O[2], batch, g, Nn, 128);
    gat_mlp<<<(G * 32 + 255) / 256, 256, 0, stream>>>(g, lin1_W, lin1_b, g1, G, 128, 32, 1);
    gat_mlp<<<(G * C + 255) / 256, 256, 0, stream>>>(g1, lin2_W, lin2_b, lg, G, 32, C, 0);
    gat_log_softmax0<<<1, 640, 0, stream>>>(lg, (float*)d_out, G, C);
}